// PagedAttentionImpl_70489003262670
// MI455X (gfx1250) — compile-verified
//
#include <hip/hip_runtime.h>

#define NUM_SEQS 32
#define NUM_HEADS 32
#define NUM_KV_HEADS 8
#define GROUP 4
#define HEAD_DIM 128
#define BLOCK_SIZE 16
#define MAX_CTX 2048
#define BLOCKS_PER_SEQ 128
#define WAVES 8
#define SCALEF 0.08838834764831845f  // 1/sqrt(128)

typedef float v2f __attribute__((ext_vector_type(2)));
typedef float v4f __attribute__((ext_vector_type(4)));
typedef float v8f __attribute__((ext_vector_type(8)));

__global__ __launch_bounds__(256)
void paged_attn_wmma_kernel(const float* __restrict__ q,
                            const float* __restrict__ k,
                            const float* __restrict__ v,
                            const float* __restrict__ k_cache,
                            const float* __restrict__ v_cache,
                            const float* __restrict__ cos_cache,
                            const float* __restrict__ sin_cache,
                            const int*   __restrict__ position,
                            const int*   __restrict__ block_tables,
                            const int*   __restrict__ context_lens,
                            float* __restrict__ out)
{
    __shared__ __align__(16) float sm_o[WAVES][GROUP][HEAD_DIM];
    __shared__ float sm_m[WAVES][GROUP];
    __shared__ float sm_l[WAVES][GROUP];
    __shared__ float sm_M[GROUP];
    __shared__ float sm_den[GROUP];

    const int kvh  = blockIdx.x;      // 0..7
    const int s    = blockIdx.y;      // 0..31
    const int tid  = threadIdx.x;
    const int w    = tid >> 5;        // wave id 0..7
    const int lane = tid & 31;
    const int hn   = lane & 15;       // A: row (head-in-group, padded) / B: token-in-tile
    const int lg   = lane >> 4;       // lane-group -> K offset 2*lg

    const int ctx = context_lens[s];
    const int pos = position[s];

    const v2f* cp = (const v2f*)(cos_cache + (size_t)pos * (HEAD_DIM / 2));
    const v2f* sp = (const v2f*)(sin_cache + (size_t)pos * (HEAD_DIM / 2));

    // ---- Q A-fragments with inline NeoX RoPE -------------------------------
    // A (16x4 f32): lane L<16 holds A[M=L][K=0..1] in {v0,v1}; L>=16 holds K=2..3.
    // Step kk covers dims 4kk+2*lg+{0,1}; dims d and d+64 (RoPE pair) map to
    // steps kk and kk+16 in the SAME lane, so RoPE is lane-local.
    v2f qa[32];
    if (hn < GROUP) {
        const v2f* qp = (const v2f*)(q + (size_t)(s * NUM_HEADS + kvh * GROUP + hn) * HEAD_DIM);
        #pragma unroll
        for (int kk = 0; kk < 16; ++kk) {
            v2f c2 = cp[2 * kk + lg], s2 = sp[2 * kk + lg];
            v2f lo = qp[2 * kk + lg], hi = qp[2 * kk + 32 + lg];
            qa[kk].x      = lo.x * c2.x - hi.x * s2.x;
            qa[kk].y      = lo.y * c2.y - hi.y * s2.y;
            qa[kk + 16].x = hi.x * c2.x + lo.x * s2.x;
            qa[kk + 16].y = hi.y * c2.y + lo.y * s2.y;
        }
    } else {
        #pragma unroll
        for (int kk = 0; kk < 32; ++kk) { qa[kk].x = 0.f; qa[kk].y = 0.f; }
    }

    float m[GROUP], l[GROUP];
    v4f   oa[GROUP];
    #pragma unroll
    for (int h = 0; h < GROUP; ++h) {
        m[h] = -1e30f; l[h] = 0.f;
        oa[h] = (v4f){0.f, 0.f, 0.f, 0.f};
    }

    // ---- flash-decode over 16-token tiles, strided across 8 waves ----------
    for (int tile = w; tile * BLOCK_SIZE < ctx; tile += WAVES) {
        const int t0 = tile * BLOCK_SIZE;
        const int t  = t0 + hn;
        const int blk = block_tables[s * BLOCKS_PER_SEQ + tile];

        // K B-fragments: B (4x16 f32): lane L holds B[K=2*lg+{0,1}][N=L&15].
        const v2f* kp = (const v2f*)(k_cache +
            (((size_t)blk * BLOCK_SIZE + hn) * NUM_KV_HEADS + kvh) * HEAD_DIM);
        v2f kb[32];
        #pragma unroll
        for (int kk = 0; kk < 32; ++kk) kb[kk] = kp[2 * kk + lg];

        // newest token: override cached K with RoPE'd fresh k (cache not mutated)
        if (t == pos) {
            const v2f* knp = (const v2f*)(k + (size_t)(s * NUM_KV_HEADS + kvh) * HEAD_DIM);
            #pragma unroll
            for (int kk = 0; kk < 16; ++kk) {
                v2f c2 = cp[2 * kk + lg], s2 = sp[2 * kk + lg];
                v2f lo = knp[2 * kk + lg], hi = knp[2 * kk + 32 + lg];
                kb[kk].x      = lo.x * c2.x - hi.x * s2.x;
                kb[kk].y      = lo.y * c2.y - hi.y * s2.y;
                kb[kk + 16].x = hi.x * c2.x + lo.x * s2.x;
                kb[kk + 16].y = hi.y * c2.y + lo.y * s2.y;
            }
        }

        // scores tile: 16x16 = (4 real heads padded to 16) x (16 tokens)
        v8f acc = {};
        #pragma unroll
        for (int kk = 0; kk < 32; ++kk)
            acc = __builtin_amdgcn_wmma_f32_16x16x4_f32(
                      false, qa[kk], false, kb[kk], (short)0, acc, false, false);

        // online softmax: C layout -> lane n<16 holds score[h][n] in acc[h], h<4
        float alpha[GROUP], p[GROUP];
        #pragma unroll
        for (int h = 0; h < GROUP; ++h) {
            float sv = (t < ctx) ? acc[h] * SCALEF : -1e30f;
            float mt = sv;
            mt = fmaxf(mt, __shfl_xor(mt, 1, 32));
            mt = fmaxf(mt, __shfl_xor(mt, 2, 32));
            mt = fmaxf(mt, __shfl_xor(mt, 4, 32));
            mt = fmaxf(mt, __shfl_xor(mt, 8, 32));
            mt = __shfl(mt, 0, 32);
            float mn = fmaxf(m[h], mt);
            alpha[h] = __expf(m[h] - mn);
            m[h] = mn;
            p[h] = __expf(sv - mn);
            float rs = p[h];
            rs += __shfl_xor(rs, 1, 32);
            rs += __shfl_xor(rs, 2, 32);
            rs += __shfl_xor(rs, 4, 32);
            rs += __shfl_xor(rs, 8, 32);
            rs = __shfl(rs, 0, 32);
            l[h] = l[h] * alpha[h] + rs;
            oa[h] *= alpha[h];
        }

        // P*V: lane owns dims 4*lane..4*lane+3; coalesced 512B row per load
        #pragma unroll
        for (int n = 0; n < BLOCK_SIZE; ++n) {
            const float* vb = (t0 + n == pos)
                ? (v + (size_t)(s * NUM_KV_HEADS + kvh) * HEAD_DIM)
                : (v_cache + (((size_t)blk * BLOCK_SIZE + n) * NUM_KV_HEADS + kvh) * HEAD_DIM);
            v4f vn = *(const v4f*)(vb + 4 * lane);
            #pragma unroll
            for (int h = 0; h < GROUP; ++h) {
                float pn = __shfl(p[h], n, 32);
                oa[h] += pn * vn;
            }
        }
    }

    // ---- cross-wave combine -------------------------------------------------
    #pragma unroll
    for (int h = 0; h < GROUP; ++h)
        *(v4f*)(&sm_o[w][h][4 * lane]) = oa[h];
    if (lane == 0) {
        #pragma unroll
        for (int h = 0; h < GROUP; ++h) { sm_m[w][h] = m[h]; sm_l[w][h] = l[h]; }
    }
    __syncthreads();

    if (tid < GROUP) {
        float M = -1e30f;
        for (int ww = 0; ww < WAVES; ++ww) M = fmaxf(M, sm_m[ww][tid]);
        float den = 0.f;
        for (int ww = 0; ww < WAVES; ++ww)
            den += __expf(sm_m[ww][tid] - M) * sm_l[ww][tid];
        sm_M[tid] = M; sm_den[tid] = den;
    }
    __syncthreads();

    if (tid < GROUP * (HEAD_DIM / 4)) {           // 128 threads: one float4 each
        const int h  = tid >> 5;
        const int dc = (tid & 31) * 4;
        v4f accv = {0.f, 0.f, 0.f, 0.f};
        for (int ww = 0; ww < WAVES; ++ww) {
            float sc = __expf(sm_m[ww][h] - sm_M[h]);
            accv += sc * (*(const v4f*)(&sm_o[ww][h][dc]));
        }
        accv *= (1.0f / sm_den[h]);
        *(v4f*)(out + (size_t)s * (NUM_HEADS * HEAD_DIM)
                    + (size_t)(kvh * GROUP + h) * HEAD_DIM + dc) = accv;
    }
}

extern "C" void kernel_launch(void* const* d_in, const int* in_sizes, int n_in,
                              void* d_out, int out_size, void* d_ws, size_t ws_size,
                              hipStream_t stream) {
    const float* q            = (const float*)d_in[0];
    const float* k            = (const float*)d_in[1];
    const float* v            = (const float*)d_in[2];
    const float* k_cache      = (const float*)d_in[3];
    const float* v_cache      = (const float*)d_in[4];
    const float* cos_cache    = (const float*)d_in[5];
    const float* sin_cache    = (const float*)d_in[6];
    const int*   position     = (const int*)d_in[7];
    // d_in[8] slot_mapping: implied by position/block_tables (identity layout)
    const int*   block_tables = (const int*)d_in[9];
    const int*   context_lens = (const int*)d_in[10];
    float* out = (float*)d_out;

    dim3 grid(NUM_KV_HEADS, NUM_SEQS);
    paged_attn_wmma_kernel<<<grid, 256, 0, stream>>>(
        q, k, v, k_cache, v_cache, cos_cache, sin_cache,
        position, block_tables, context_lens, out);
}